// CausalSelfAttention_25477745999965
// MI455X (gfx1250) — compile-verified
//
#include <hip/hip_runtime.h>

// MI455X / gfx1250 fused windowed-causal self-attention, fp32 end-to-end.
// All GEMMs use V_WMMA_F32_16X16X4_F32 (fp32 matrix pipe, matches reference
// precision). Attention K/V tiles are staged once per 128-query block into
// double-buffered LDS via GLOBAL_LOAD_ASYNC_TO_LDS_B128 (ASYNCcnt): the DMA
// of tile i+1 overlaps the WMMA/softmax compute of tile i, with a single
// block barrier per tile.

typedef float v2f __attribute__((ext_vector_type(2)));
typedef float v8f __attribute__((ext_vector_type(8)));

#define B_DIM 2
#define C_DIM 256
#define T_DIM 4096
#define H_DIM 4
#define D_DIM 64
#define WIN   1024

__device__ __forceinline__ v8f wmma_f32(v2f a, v2f b, v8f c) {
  // D = A(16x4 f32) * B(4x16 f32) + C(16x16 f32)
  return __builtin_amdgcn_wmma_f32_16x16x4_f32(
      /*neg_a=*/false, a, /*neg_b=*/false, b,
      /*c_mod=*/(short)0, c, /*reuse_a=*/false, /*reuse_b=*/false);
}

// Per-lane async 16B copy global -> LDS, tracked by ASYNCcnt.
__device__ __forceinline__ void async_copy_b128(unsigned lds_addr,
                                                const float* gptr) {
  unsigned long long ga = (unsigned long long)(size_t)gptr;
  asm volatile("global_load_async_to_lds_b128 %0, %1, off"
               :: "v"(lds_addr), "v"(ga)
               : "memory");
}
__device__ __forceinline__ void wait_async0() {
  asm volatile("s_wait_asynccnt 0" ::: "memory");
}

// ---------------------------------------------------------------------------
// Kernel 1: qkv = W_qkv @ x + b_qkv   ([768x256] x [256x4096] per batch)
// One wave computes one 16(o) x 16(t) tile, K-loop over 256 channels step 4.
// Q stored [B,H,T,D] (pre-scaled by 1/sqrt(D)), K stored [B,H,D,T],
// V stored [B,H,T,D].
// ---------------------------------------------------------------------------
__global__ __launch_bounds__(256) void qkv_kernel(
    const float* __restrict__ x, const float* __restrict__ Wqkv,
    const float* __restrict__ bqkv, float* __restrict__ qws,
    float* __restrict__ kws, float* __restrict__ vws) {
  const int lane = threadIdx.x & 31;
  const int wave = threadIdx.x >> 5;
  const int lo = lane & 15;   // row (A) / col (B,C)
  const int hi = lane >> 4;   // half-wave select
  const int nT = T_DIM / 16;  // 256
  const int nO = (3 * C_DIM) / 16;  // 48
  int tile = blockIdx.x * 8 + wave;     // [0, B*48*256)
  int b  = tile / (nO * nT);
  int r  = tile % (nO * nT);
  int o0 = (r / nT) * 16;
  int t0 = (r % nT) * 16;

  const float* wrow = Wqkv + (size_t)(o0 + lo) * C_DIM + 2 * hi;
  const float* xcol = x + (size_t)b * C_DIM * T_DIM + t0 + lo;

  v8f acc = {0, 0, 0, 0, 0, 0, 0, 0};
  for (int c0 = 0; c0 < C_DIM; c0 += 4) {
    v2f a = *(const v2f*)(wrow + c0);               // A[m=lo][k=2hi,2hi+1]
    v2f bb;                                          // B[k=j+2hi][n=lo]
    bb.x = xcol[(size_t)(c0 + 0 + 2 * hi) * T_DIM];
    bb.y = xcol[(size_t)(c0 + 1 + 2 * hi) * T_DIM];
    acc = wmma_f32(a, bb, acc);
  }

  const int which = o0 / C_DIM;       // 0=q,1=k,2=v (tile never straddles)
  const int oin   = o0 % C_DIM;
  const int h     = oin / D_DIM;
  const int d0    = oin % D_DIM;
  const int bh    = b * H_DIM + h;
  const int t     = t0 + lo;
#pragma unroll
  for (int v = 0; v < 8; ++v) {
    int row = v + 8 * hi;
    float val = acc[v] + bqkv[o0 + row];
    int d = d0 + row;
    if (which == 0)
      qws[((size_t)bh * T_DIM + t) * D_DIM + d] = val * 0.125f;  // 1/sqrt(64)
    else if (which == 1)
      kws[((size_t)bh * D_DIM + d) * T_DIM + t] = val;
    else
      vws[((size_t)bh * T_DIM + t) * D_DIM + d] = val;
  }
}

// ---------------------------------------------------------------------------
// Kernel 2: flash attention. One block = 128 consecutive queries of one
// (b,h); 8 waves x 16 queries. Double-buffered async K/V staging:
//   wait(pair i); barrier; issue(pair i+1, other buffer); compute(tile i).
// Each in-window wave runs 16 S-WMMAs + fp32 online softmax + LDS
// P-transpose + 16 PV-WMMAs, all operands served from LDS.
// ---------------------------------------------------------------------------
__global__ __launch_bounds__(256) void attn_kernel(
    const float* __restrict__ qws, const float* __restrict__ kws,
    const float* __restrict__ vws, float* __restrict__ yws) {
  __shared__ __align__(16) float Ktile[2][64][16];   // [buf][d][key]
  __shared__ __align__(16) float Vtile[2][16][72];   // [buf][key][d] (pad 72)
  __shared__ __align__(16) float pbuf[8][16][18];    // per-wave P transpose

  const int tid  = threadIdx.x;
  const int lane = tid & 31;
  const int wave = tid >> 5;
  const int lo = lane & 15;
  const int hi = lane >> 4;

  int blk  = blockIdx.x;          // [0, B*H*32)
  int bh   = blk >> 5;            // (b,h)
  int tblk = (blk & 31) * 128;    // block's first query
  int t0   = tblk + wave * 16;    // this wave's query tile

  // Preload Q A-fragments: lane lo holds query row t0+lo, cols 4j+2hi+{0,1}.
  const float* qrow = qws + ((size_t)bh * T_DIM + t0 + lo) * D_DIM + 2 * hi;
  v2f qa[16];
#pragma unroll
  for (int j = 0; j < 16; ++j) qa[j] = *(const v2f*)(qrow + 4 * j);

  float mrow[8], lrow[8];
  v8f oacc[4];
#pragma unroll
  for (int v = 0; v < 8; ++v) { mrow[v] = -__builtin_inff(); lrow[v] = 0.f; }
#pragma unroll
  for (int dt = 0; dt < 4; ++dt) oacc[dt] = (v8f){0, 0, 0, 0, 0, 0, 0, 0};

  const float* kbase = kws + (size_t)bh * D_DIM * T_DIM;  // [D][T]
  const float* vbase = vws + (size_t)bh * T_DIM * D_DIM;  // [T][D]

  // Cooperative copy slots: 256 threads x 16B cover each 4KB tile.
  const int kd  = tid >> 2;            // 0..63  (d row of K tile)
  const int kn4 = (tid & 3) * 4;       // 0,4,8,12 (key col group)
  const int vk  = tid >> 4;            // 0..15  (key row of V tile)
  const int vd4 = (tid & 15) * 4;      // 0..60  (d col group)
  const unsigned kdst[2] = {(unsigned)(size_t)&Ktile[0][kd][kn4],
                            (unsigned)(size_t)&Ktile[1][kd][kn4]};
  const unsigned vdst[2] = {(unsigned)(size_t)&Vtile[0][vk][vd4],
                            (unsigned)(size_t)&Vtile[1][vk][vd4]};
  const float* kgs = kbase + (size_t)kd * T_DIM + kn4;
  const float* vgs = vbase + (size_t)vk * D_DIM + vd4;

  const int klo = (tblk >= WIN) ? (tblk - WIN) : 0;
  const int khi = tblk + 112;  // last wave's diagonal tile
  const int ntiles = ((khi - klo) >> 4) + 1;

  // Prologue: stage tile 0 into buffer 0.
  async_copy_b128(kdst[0], kgs + klo);
  async_copy_b128(vdst[0], vgs + (size_t)klo * D_DIM);

  for (int i = 0; i < ntiles; ++i) {
    const int kt0 = klo + (i << 4);
    const int p   = i & 1;

    wait_async0();      // this wave's chunks of tile i landed in LDS
    __syncthreads();    // all waves' chunks of tile i visible

    // Prefetch tile i+1 into the other buffer; its previous reader
    // (compute of tile i-1) finished before the barrier above.
    if (i + 1 < ntiles) {
      const int kt1 = kt0 + 16;
      const int pn  = p ^ 1;
      async_copy_b128(kdst[pn], kgs + kt1);
      async_copy_b128(vdst[pn], vgs + (size_t)kt1 * D_DIM);
    }

    // Wave-uniform window test (keeps EXEC all-ones for WMMA).
    bool active = (kt0 <= t0) && (kt0 + WIN >= t0);
    if (active) {
      // ---- S = (Q^T)(K): 16 f32 WMMAs, K-dim 64 in steps of 4 ----
      v8f s = {0, 0, 0, 0, 0, 0, 0, 0};
#pragma unroll
      for (int j = 0; j < 16; ++j) {
        v2f kb;  // B[k=j'+2hi][n=lo] = K[d=4j+j'+2hi][kt0+lo]
        kb.x = Ktile[p][4 * j + 0 + 2 * hi][lo];
        kb.y = Ktile[p][4 * j + 1 + 2 * hi][lo];
        s = wmma_f32(qa[j], kb, s);
      }

      // ---- mask + online softmax (row v+8hi lives across a 16-lane half)
      const int skey = kt0 + lo;
#pragma unroll
      for (int v = 0; v < 8; ++v) {
        const int tq = t0 + v + 8 * hi;
        bool ok = (skey <= tq) && (skey + WIN > tq);
        float sv = ok ? s[v] : -__builtin_inff();
        float rm = sv;
        rm = fmaxf(rm, __shfl_xor(rm, 1));
        rm = fmaxf(rm, __shfl_xor(rm, 2));
        rm = fmaxf(rm, __shfl_xor(rm, 4));
        rm = fmaxf(rm, __shfl_xor(rm, 8));
        float mn = fmaxf(mrow[v], rm);
        bool live = (mn > -__builtin_inff());
        float scale = live ? __expf(mrow[v] - mn) : 1.0f;
        float pexp  = live ? __expf(sv - mn) : 0.0f;
        float rs = pexp;
        rs += __shfl_xor(rs, 1);
        rs += __shfl_xor(rs, 2);
        rs += __shfl_xor(rs, 4);
        rs += __shfl_xor(rs, 8);
        lrow[v] = lrow[v] * scale + rs;
        mrow[v] = mn;
#pragma unroll
        for (int dt = 0; dt < 4; ++dt) oacc[dt][v] *= scale;
        pbuf[wave][v + 8 * hi][lo] = pexp;  // C-layout -> LDS (row, col)
      }

      // ---- O += P*V (A = P via LDS transpose, B = V tile from LDS) ----
#pragma unroll
      for (int dt = 0; dt < 4; ++dt) {
#pragma unroll
        for (int kk = 0; kk < 4; ++kk) {
          v2f pa = *(const v2f*)(&pbuf[wave][lo][4 * kk + 2 * hi]);
          v2f vb;  // B[k=j'+2hi][n=lo] = V[key=4kk+j'+2hi][dt*16+lo]
          vb.x = Vtile[p][4 * kk + 0 + 2 * hi][dt * 16 + lo];
          vb.y = Vtile[p][4 * kk + 1 + 2 * hi][dt * 16 + lo];
          oacc[dt] = wmma_f32(pa, vb, oacc[dt]);
        }
      }
    }
  }

  // ---- normalize and store y as [B,H,T,D]
#pragma unroll
  for (int v = 0; v < 8; ++v) {
    float inv = 1.0f / lrow[v];  // diagonal key always valid -> lrow > 0
    int tq = t0 + v + 8 * hi;
#pragma unroll
    for (int dt = 0; dt < 4; ++dt)
      yws[((size_t)bh * T_DIM + tq) * D_DIM + dt * 16 + lo] = oacc[dt][v] * inv;
  }
}

// ---------------------------------------------------------------------------
// Kernel 3: out = x + W_proj @ y + b_proj  ([256x256] x [256x4096] per batch)
// ---------------------------------------------------------------------------
__global__ __launch_bounds__(256) void proj_kernel(
    const float* __restrict__ x, const float* __restrict__ Wproj,
    const float* __restrict__ bproj, const float* __restrict__ yws,
    float* __restrict__ out) {
  const int lane = threadIdx.x & 31;
  const int wave = threadIdx.x >> 5;
  const int lo = lane & 15;
  const int hi = lane >> 4;
  const int nT = T_DIM / 16;   // 256
  const int nO = C_DIM / 16;   // 16
  int tile = blockIdx.x * 8 + wave;  // [0, B*16*256)
  int b  = tile / (nO * nT);
  int r  = tile % (nO * nT);
  int o0 = (r / nT) * 16;
  int t0 = (r % nT) * 16;

  const float* wrow = Wproj + (size_t)(o0 + lo) * C_DIM + 2 * hi;
  v8f acc = {0, 0, 0, 0, 0, 0, 0, 0};
  for (int c0 = 0; c0 < C_DIM; c0 += 4) {
    v2f a = *(const v2f*)(wrow + c0);
    int h = c0 / D_DIM;            // 4-chunk never straddles a head
    int d = (c0 % D_DIM) + 2 * hi; // consecutive d -> b64 load from y[T][D]
    v2f bb = *(const v2f*)(yws +
        (((size_t)(b * H_DIM + h) * T_DIM) + t0 + lo) * D_DIM + d);
    acc = wmma_f32(a, bb, acc);
  }

#pragma unroll
  for (int v = 0; v < 8; ++v) {
    int o = o0 + v + 8 * hi;
    size_t idx = ((size_t)b * C_DIM + o) * T_DIM + t0 + lo;
    out[idx] = acc[v] + bproj[o] + x[idx];
  }
}

// ---------------------------------------------------------------------------
extern "C" void kernel_launch(void* const* d_in, const int* in_sizes, int n_in,
                              void* d_out, int out_size, void* d_ws,
                              size_t ws_size, hipStream_t stream) {
  const float* x     = (const float*)d_in[0];  // [2,256,4096]
  const float* Wqkv  = (const float*)d_in[1];  // [768,256]
  const float* bqkv  = (const float*)d_in[2];  // [768]
  const float* Wproj = (const float*)d_in[3];  // [256,256]
  const float* bproj = (const float*)d_in[4];  // [256]
  float* out = (float*)d_out;                  // [2,256,4096]

  // Workspace: q | k | v | y, each B*H*T*D = 2,097,152 floats (32 MB total).
  const size_t SEG = (size_t)B_DIM * H_DIM * T_DIM * D_DIM;
  float* qws = (float*)d_ws;
  float* kws = qws + SEG;
  float* vws = kws + SEG;
  float* yws = vws + SEG;

  // 1) QKV projection: B*48*256 = 24576 wave-tiles, 8 waves/block.
  qkv_kernel<<<24576 / 8, 256, 0, stream>>>(x, Wqkv, bqkv, qws, kws, vws);
  // 2) Windowed-causal flash attention: one block per 128 queries of (b,h).
  attn_kernel<<<B_DIM * H_DIM * (T_DIM / 128), 256, 0, stream>>>(qws, kws, vws,
                                                                 yws);
  // 3) Output projection + residual: B*16*256 = 8192 wave-tiles.
  proj_kernel<<<8192 / 8, 256, 0, stream>>>(x, Wproj, bproj, yws, out);
}